// GNN_Loss_84945863180679
// MI455X (gfx1250) — compile-verified
//
#include <hip/hip_runtime.h>
#include <math.h>

typedef __attribute__((ext_vector_type(2))) float v2f;
typedef __attribute__((ext_vector_type(8))) float v8f;

#define HH 20
#define CC 36
#define STEPLEN 2.0f
#define OBS_RADIUS 1.0f
#define VEH_R2 4.0f   // VEH_RADIUS^2
#define TWOPI 6.28318530717958647692f

__device__ __forceinline__ float waveSum32(float x) {
  #pragma unroll
  for (int off = 16; off; off >>= 1) x += __shfl_xor(x, off, 32);
  return x;
}

// ---------------------------------------------------------------------------
// K0: zero the 6 scalar accumulators (ws is poisoned, must be cleared per call)
__global__ void k_init(float* __restrict__ acc) {
  if (threadIdx.x < 16) acc[threadIdx.x] = 0.0f;
}

// ---------------------------------------------------------------------------
// K1: per-(n,h) class decode. Each wave reduces 32 consecutive logit rows
// (lane c covers the 36 classes, shuffle max+argmax), parks row k's angle in
// lane k, then all 32 lanes run sincos in parallel and store coalesced float2.
__global__ void k_decode(const float* __restrict__ pred,
                         float2* __restrict__ vecbuf, int NH) {
  const int tid  = threadIdx.x;
  const int lane = tid & 31;
  const int w    = tid >> 5;                    // wave id within block
  for (int base = blockIdx.x * blockDim.x; base < NH;
       base += gridDim.x * blockDim.x) {
    const int wbase = base + w * 32;            // this wave's 32 items
    float myAng = 0.0f;
    #pragma unroll 1
    for (int k = 0; k < 32; ++k) {
      int item = wbase + k;                     // wave-uniform
      float ang = 0.0f;
      if (item < NH) {
        const float* bp = pred + (size_t)item * CC;
        float val = bp[lane];                   // classes 0..31
        int   idx = lane;
        if (lane < CC - 32) {                   // classes 32..35
          float v2 = bp[lane + 32];
          if (v2 > val) { val = v2; idx = lane + 32; }  // strict > = first-max
        }
        #pragma unroll
        for (int off = 16; off; off >>= 1) {    // wave32 max+argmax
          float ov = __shfl_xor(val, off, 32);
          int   oi = __shfl_xor(idx, off, 32);
          if (ov > val || (ov == val && oi < idx)) { val = ov; idx = oi; }
        }
        ang = (float)idx * (TWOPI / (float)CC) * val;
      }
      if (lane == k) myAng = ang;               // transpose into lane k
    }
    int item = wbase + lane;
    if (item < NH) {
      float s, c;
      sincosf(myAng, &s, &c);
      vecbuf[item] = make_float2(c * STEPLEN, s * STEPLEN);  // coalesced
    }
  }
}

// ---------------------------------------------------------------------------
// K2: per-node rollout (inclusive prefix-sum over H via shfl_up), overwrite
// the vec buffer with trajectory points, and accumulate the distance term.
__global__ void k_rollout(const float* __restrict__ X,
                          float2* __restrict__ traj,
                          float* __restrict__ acc, int N) {
  const int lane = threadIdx.x & 31;
  const int wpb  = blockDim.x >> 5;
  int gw      = blockIdx.x * wpb + (threadIdx.x >> 5);
  int wstride = gridDim.x * wpb;
  float distAcc = 0.0f, nvehAcc = 0.0f;
  for (int n = gw; n < N; n += wstride) {
    const float* xr = X + (size_t)n * 5;
    float tp = xr[0], px = xr[1], py = xr[2], tx = xr[3], ty = xr[4];
    bool veh = (tp == 0.0f);
    float sx = 0.0f, sy = 0.0f;
    if (lane < HH) { float2 v = traj[n * HH + lane]; sx = v.x; sy = v.y; }
    #pragma unroll
    for (int off = 1; off < 32; off <<= 1) {         // inclusive scan
      float ox = __shfl_up(sx, off, 32);
      float oy = __shfl_up(sy, off, 32);
      if (lane >= off) { sx += ox; sy += oy; }
    }
    float trx = veh ? (px + sx) : 0.0f;
    float trr = veh ? (py + sy) : 0.0f;
    if (lane < HH) {
      traj[n * HH + lane] = make_float2(trx, trr);
      if (veh) {
        float dx = tx - trx, dy = ty - trr;
        distAcc += sqrtf(dx * dx + dy * dy);
      }
    }
    if (lane == 0 && veh) nvehAcc += 1.0f;
  }
  distAcc = waveSum32(distAcc);
  if (lane == 0) {
    atomicAdd(&acc[0], distAcc);
    atomicAdd(&acc[1], nvehAcc);
  }
}

// ---------------------------------------------------------------------------
// K3: obstacle term. One wave per edge; lanes 0..19 cover the horizon.
// Branchless: d^2 < thr^2 (thr>0 guard) and v_rsq_f32 for 1/od.
__global__ void k_obstacle(const float* __restrict__ X,
                           const float2* __restrict__ traj,
                           const int* __restrict__ e0,
                           const int* __restrict__ e1,
                           float* __restrict__ acc, int MO) {
  const int lane = threadIdx.x & 31;
  const int wpb  = blockDim.x >> 5;
  int gw      = blockIdx.x * wpb + (threadIdx.x >> 5);
  int wstride = gridDim.x * wpb;
  float sumL = 0.0f, cntL = 0.0f;
  for (int e = gw; e < MO; e += wstride) {
    int o = e0[e], t = e1[e];
    const float* xr = X + (size_t)o * 5;
    float ox = xr[1], oy = xr[2];
    float thr  = xr[3] + OBS_RADIUS;
    float thr2 = (thr > 0.0f) ? thr * thr : -1.0f;   // <0 => never near
    if (lane < HH) {
      float2 tt = traj[t * HH + lane];
      float dx = ox - tt.x, dy = oy - tt.y;
      float d2 = dx * dx + dy * dy;
      float r  = __builtin_amdgcn_rsqf(fmaxf(d2, 1e-24f));  // 1/od
      bool nr  = d2 < thr2;
      cntL += nr ? 1.0f : 0.0f;
      sumL += nr ? r : 0.0f;
    }
  }
  sumL = waveSum32(sumL);
  cntL = waveSum32(cntL);
  if (lane == 0) {
    atomicAdd(&acc[2], sumL);
    atomicAdd(&acc[3], cntL);
  }
}

// ---------------------------------------------------------------------------
// K4: vehicle term. One wave per edge. 16x16 core of the 20x20 pairwise-d^2
// matrix via V_WMMA_F32_16X16X4_F32 rank-4 Gram trick:
//   A[i,:] = [x_i, y_i, |p_i|^2, 1],  B[:,j] = [-2x_j, -2y_j, 1, |q_j|^2]
//   (A·B)[i,j] = |p_i - q_j|^2
// 144 border pairs (i>=16 or j>=16) handled with VALU. Branchless rsq sums.
__global__ void k_vehicle(const float2* __restrict__ traj,
                          const int* __restrict__ e0,
                          const int* __restrict__ e1,
                          float* __restrict__ acc, int MV) {
  const int lane = threadIdx.x & 31;
  const int wpb  = blockDim.x >> 5;
  int gw      = blockIdx.x * wpb + (threadIdx.x >> 5);
  int wstride = gridDim.x * wpb;
  float sumL = 0.0f, cntL = 0.0f;
  for (int e = gw; e < MV; e += wstride) {   // e is wave-uniform -> EXEC all-1s
    int a = e0[e], b = e1[e];
    const float2* fa = traj + (size_t)a * HH;
    const float2* fb = traj + (size_t)b * HH;
    int  m  = lane & 15;
    bool hi = (lane >= 16);
    float2 fm = fa[m];
    float2 tn = fb[m];
    float sm  = fm.x * fm.x + fm.y * fm.y;   // |p_m|^2
    float tq  = tn.x * tn.x + tn.y * tn.y;   // |q_n|^2
    // 16x4 f32 A: lanes 0-15 hold K=0,1; lanes 16-31 hold K=2,3 (per row m)
    v2f A, B;
    A[0] = hi ? sm   : fm.x;
    A[1] = hi ? 1.0f : fm.y;
    // 4x16 f32 B: lanes 0-15 hold K=0,1; lanes 16-31 hold K=2,3 (per col n)
    B[0] = hi ? 1.0f : (-2.0f * tn.x);
    B[1] = hi ? tq   : (-2.0f * tn.y);
    v8f Cz = {0.f, 0.f, 0.f, 0.f, 0.f, 0.f, 0.f, 0.f};
    v8f D = __builtin_amdgcn_wmma_f32_16x16x4_f32(
        /*neg_a=*/false, A, /*neg_b=*/false, B,
        /*c_mod=*/(short)0, Cz, /*reuse_a=*/false, /*reuse_b=*/false);
    // every accumulator entry is a valid (i<16, j<16) pair -> no masking
    #pragma unroll
    for (int k = 0; k < 8; ++k) {
      float d2 = D[k];
      float r  = __builtin_amdgcn_rsqf(fmaxf(d2, 1e-24f));
      bool nr  = d2 < VEH_R2;
      cntL += nr ? 1.0f : 0.0f;
      sumL += nr ? r : 0.0f;
    }
    // border: i in [16,20) x j in [0,20)  (80 pairs), then
    //         i in [0,16)  x j in [16,20) (64 pairs)  => 144 total
    #pragma unroll
    for (int it = 0; it < 5; ++it) {
      int p = lane + it * 32;
      if (p < 144) {
        int i, j;
        if (p < 80) { i = 16 + p / 20; j = p % 20; }
        else        { int q = p - 80; i = q >> 2; j = 16 + (q & 3); }
        float2 fi = fa[i], tj = fb[j];
        float dx = fi.x - tj.x, dy = fi.y - tj.y;
        float d2 = dx * dx + dy * dy;
        float r  = __builtin_amdgcn_rsqf(fmaxf(d2, 1e-24f));
        bool nr  = d2 < VEH_R2;
        cntL += nr ? 1.0f : 0.0f;
        sumL += nr ? r : 0.0f;
      }
    }
  }
  sumL = waveSum32(sumL);
  cntL = waveSum32(cntL);
  if (lane == 0) {
    atomicAdd(&acc[4], sumL);
    atomicAdd(&acc[5], cntL);
  }
}

// ---------------------------------------------------------------------------
// K5: combine partials into the 4 outputs (total, dist, obstacle, vehicle).
__global__ void k_final(const float* __restrict__ acc, float* __restrict__ out) {
  if (threadIdx.x == 0) {
    float dl = acc[0] / (acc[1] * (float)HH);                     // DIST_COST=1
    float ol = (acc[3] > 0.0f) ? acc[2] / fmaxf(acc[3], 1.0f) : 0.0f;
    float vl = (acc[5] > 0.0f) ? acc[4] / fmaxf(acc[5], 1.0f) : 0.0f;
    out[0] = dl + ol + vl;
    out[1] = dl;
    out[2] = ol;
    out[3] = vl;
  }
}

// ---------------------------------------------------------------------------
extern "C" void kernel_launch(void* const* d_in, const int* in_sizes, int n_in,
                              void* d_out, int out_size, void* d_ws, size_t ws_size,
                              hipStream_t stream) {
  (void)n_in; (void)out_size; (void)ws_size;
  const float* X    = (const float*)d_in[0];   // (N,5) f32
  const float* pred = (const float*)d_in[1];   // (N,H*C) f32
  const int*   ev   = (const int*)d_in[2];     // (2,MV) i32
  const int*   eo   = (const int*)d_in[3];     // (2,MO) i32
  int N  = in_sizes[0] / 5;
  int MV = in_sizes[2] / 2;
  int MO = in_sizes[3] / 2;
  int NH = N * HH;

  float*  ws   = (float*)d_ws;
  float*  acc  = ws;                     // 6 accumulators (16 floats reserved)
  float2* traj = (float2*)(ws + 16);     // N*H float2: vec scratch then traj

  k_init<<<1, 32, 0, stream>>>(acc);
  k_decode<<<4096, 256, 0, stream>>>(pred, traj, NH);
  k_rollout<<<1024, 256, 0, stream>>>(X, traj, acc, N);
  k_obstacle<<<2048, 256, 0, stream>>>(X, traj, eo, eo + MO, acc, MO);
  k_vehicle<<<2048, 256, 0, stream>>>(traj, ev, ev + MV, acc, MV);
  k_final<<<1, 32, 0, stream>>>(acc, (float*)d_out);
}